// GNN_29661044146285
// MI455X (gfx1250) — compile-verified
//
#include <hip/hip_runtime.h>
#include <math.h>

typedef float v2f __attribute__((ext_vector_type(2)));
typedef float v8f __attribute__((ext_vector_type(8)));

#define GB 8
#define GN 4096
#define GE 16384
#define GK 16
#define GF 128

// ---------------------------------------------------------------------------
// Copy x0 (B,N,128) into out (B,N,384) at column offset 0.  float4 coalesced.
// ---------------------------------------------------------------------------
__global__ void gnn_copy_x0(const float* __restrict__ x0, float* __restrict__ out) {
    size_t i   = (size_t)blockIdx.x * blockDim.x + threadIdx.x;   // over B*N*(F/4)
    size_t row = i >> 5;                                          // 32 float4 per row
    int    c   = (int)(i & 31);
    float4 v = ((const float4*)(x0 + row * GF))[c];
    ((float4*)(out + row * (3 * GF)))[c] = v;
}

// ---------------------------------------------------------------------------
// Edge kernel: edge_emb[e,n] = tanh( b[n] + sum_k src[sender[e]][k] * W[n][k] )
// One wave per 16-edge x 128-feature tile, fp32 WMMA 16x16x4, K=128 in 32 steps.
//   A (16x4 fp32):  lane l -> row (l&15), K pair offset 2*(l>=16)   (float2 load)
//   B (4x16 fp32):  from LDS, W staged transposed once per block:
//                   Wt[k*128+n] = W[n][k]; lane l reads col n=col*16+(l&15)
//                   -> consecutive banks across lanes (conflict-light ds loads)
//   C/D (16x16):    VGPR r -> row r + 8*(l>=16), col (l&15)
// ---------------------------------------------------------------------------
__global__ void gnn_edge_wmma(const float* __restrict__ src,   // node embeds base
                              int srcStride, int srcOff,       // 128/0 (pass1), 384/128 (pass2)
                              const float* __restrict__ Wm,    // (F,F): W[n][k]
                              const float* __restrict__ bias,  // (F,)
                              const int*   __restrict__ edges, // (B,E,3) int32
                              float*       __restrict__ edge_out, // (nb,E,F) in ws
                              int b0) {
    __shared__ float Wt[GF * GF];        // 64 KB: Wt[k*128 + n] = W[n][k]

    const int lane = threadIdx.x;
    const int half = lane >> 4;          // 0: K pair {0,1}, 1: K pair {2,3}
    const int l15  = lane & 15;
    const int bb   = blockIdx.y;
    const int b    = b0 + bb;
    const int tile = blockIdx.x * blockDim.y + threadIdx.y;
    const int e0   = tile * 16;

    // Stage W -> LDS (transposed). Global reads coalesced over k.
    {
        const int tid = threadIdx.y * 32 + threadIdx.x;
        for (int idx = tid; idx < GF * GF; idx += 256) {
            const int n = idx >> 7;      // W row
            const int k = idx & 127;     // W col (contiguous across lanes)
            Wt[k * GF + n] = Wm[idx];
        }
    }
    __syncthreads();

    // sender row gather for this lane's A rows
    const int srow = edges[((size_t)b * GE + (size_t)(e0 + l15)) * 3];
    const float* __restrict__ arow =
        src + ((size_t)b * GN + (size_t)srow) * srcStride + srcOff + half * 2;

    // Preload all 32 A k-step fragments (64 VGPRs)
    v2f a[32];
#pragma unroll
    for (int k = 0; k < 32; ++k) {
        const float2 t = *(const float2*)(arow + k * 4);
        a[k].x = t.x; a[k].y = t.y;
    }

    float* __restrict__ obase = edge_out + (size_t)bb * GE * GF;

    for (int col = 0; col < 8; ++col) {
        const int   n  = col * 16 + l15;
        const float bv = bias[n];
        // Wt element for kstep k: (4k + 2*half + {0,1})*128 + n
        const float* __restrict__ bcol = Wt + n + half * (2 * GF);
        v8f acc = {bv, bv, bv, bv, bv, bv, bv, bv};
#pragma unroll
        for (int k = 0; k < 32; ++k) {
            v2f bf;
            bf.x = bcol[k * (4 * GF)];
            bf.y = bcol[k * (4 * GF) + GF];
            acc = __builtin_amdgcn_wmma_f32_16x16x4_f32(
                false, a[k], false, bf, (short)0, acc, false, false);
        }
#pragma unroll
        for (int r = 0; r < 8; ++r) {
            const int m = r + half * 8;
            obase[(size_t)(e0 + m) * GF + n] = tanhf(acc[r]);
        }
    }
}

// ---------------------------------------------------------------------------
// Node kernel: masked mean over K=16 neighbor edge embeddings.
// One wave per node; lane owns 4 contiguous features (float4, fully coalesced).
// Mask is uniform per node -> uniform branch skips dead gathers (saves HBM BW).
// ---------------------------------------------------------------------------
__global__ void gnn_node_agg(const float* __restrict__ edge_emb,   // (nb,E,F)
                             const int*   __restrict__ node_edges, // (B,N,K)
                             const int*   __restrict__ mask,       // (B,N,K)
                             float*       __restrict__ out,        // (B,N,384)
                             int outOff,                            // 128 or 256
                             int b0) {
    const int lane = threadIdx.x;
    const int bb   = blockIdx.y;
    const int b    = b0 + bb;
    const int node = blockIdx.x * blockDim.y + threadIdx.y;
    const size_t kbase = ((size_t)b * GN + (size_t)node) * GK;
    const float* __restrict__ ee = edge_emb + (size_t)bb * GE * GF;

    float4 acc = {0.f, 0.f, 0.f, 0.f};
    float  cnt = 0.f;
#pragma unroll
    for (int j = 0; j < GK; ++j) {
        const int m = mask[kbase + j];           // uniform across wave
        const int e = node_edges[kbase + j];
        if (m) {
            const float4 v = *(const float4*)(ee + (size_t)e * GF + lane * 4);
            acc.x += v.x; acc.y += v.y; acc.z += v.z; acc.w += v.w;
            cnt += 1.f;
        }
    }
    const float inv = 1.f / (cnt + 1e-8f);
    acc.x *= inv; acc.y *= inv; acc.z *= inv; acc.w *= inv;
    *(float4*)(out + ((size_t)b * GN + (size_t)node) * (3 * GF) + outOff + lane * 4) = acc;
}

// ---------------------------------------------------------------------------
extern "C" void kernel_launch(void* const* d_in, const int* in_sizes, int n_in,
                              void* d_out, int out_size, void* d_ws, size_t ws_size,
                              hipStream_t stream) {
    const float* x0         = (const float*)d_in[0];
    const int*   edges      = (const int*)  d_in[1];
    const int*   node_edges = (const int*)  d_in[2];
    const int*   mask       = (const int*)  d_in[3];
    const float* W1         = (const float*)d_in[4];
    const float* b1         = (const float*)d_in[5];
    const float* W2         = (const float*)d_in[6];
    const float* b2         = (const float*)d_in[7];
    float*       out        = (float*)d_out;
    float*       ws         = (float*)d_ws;

    // x0 -> out columns [0,128)
    {
        const size_t total = (size_t)GB * GN * (GF / 4);
        gnn_copy_x0<<<dim3((unsigned)(total / 256)), dim3(256), 0, stream>>>(x0, out);
    }

    // edge_emb scratch: full-batch if ws allows, else per-batch chunks
    const size_t fullBytes = (size_t)GB * GE * GF * sizeof(float);
    const int chunk = (ws_size >= fullBytes) ? GB : 1;

    for (int pass = 0; pass < 2; ++pass) {
        const float* src     = (pass == 0) ? x0 : out;
        const int    sStride = (pass == 0) ? GF : 3 * GF;
        const int    sOff    = (pass == 0) ? 0  : GF;
        const float* Wm      = (pass == 0) ? W1 : W2;
        const float* bs      = (pass == 0) ? b1 : b2;
        const int    outOff  = (pass == 0) ? GF : 2 * GF;

        for (int bstart = 0; bstart < GB; bstart += chunk) {
            dim3 eg(GE / 16 / 8, chunk);   // 1024 tiles / 8 waves per block
            gnn_edge_wmma<<<eg, dim3(32, 8), 0, stream>>>(
                src, sStride, sOff, Wm, bs, edges, ws, bstart);

            dim3 ng(GN / 8, chunk);        // one wave per node
            gnn_node_agg<<<ng, dim3(32, 8), 0, stream>>>(
                ws, node_edges, mask, out, outOff, bstart);
        }
    }
}